// CrossDeformableAttention_67233418052171
// MI455X (gfx1250) — compile-verified
//
#include <hip/hip_runtime.h>

// ---------------------------------------------------------------------------
// CrossDeformableAttention forward for MI455X (gfx1250, wave32).
// All dense linear algebra runs through V_WMMA_F32_16X16X4_F32 (exact f32).
// Attention is flash-style (online softmax) so the 4x4096x4096 score matrix
// is never materialized.  Workspace requirement: ~50 MB.
// ---------------------------------------------------------------------------

typedef __attribute__((ext_vector_type(2))) float v2f;
typedef __attribute__((ext_vector_type(8))) float v8f;

#define ACT_NONE 0
#define ACT_LEAKY 1
#define ACT_GELU 2
#define ACT_TANH 3

__device__ __forceinline__ float apply_act(float x, int act) {
  if (act == ACT_LEAKY) return x >= 0.0f ? x : 0.01f * x;
  if (act == ACT_GELU)  return 0.5f * x * (1.0f + erff(x * 0.70710678118654752f));
  if (act == ACT_TANH)  return tanhf(x);
  return x;
}

// ---------------------------------------------------------------------------
// Batched GEMM: C[b][m][n] = act( sum_k A[b][m][k] * opB[b][k][n] + bias[n] )
// One wave per 16x16 output tile, K-loop in steps of 4 via WMMA f32 16x16x4.
// A-frag: lane(h,lr) holds A[m0+lr][k0+2h .. +1]      (ISA 7.12.2, 32-bit A)
// B-frag: lane(h,lr) holds B[k0+2h .. +1][n0+lr]      (row-striped B)
// C/D   : lane(h,lr) VGPR r holds C[m0+r+8h][n0+lr]
// Out-of-range rows/cols are CLAMPED (not guarded): garbage lanes compute
// garbage that is never stored, keeping the hot loop branch-free.
// ---------------------------------------------------------------------------
template <int TRANSB>
__global__ void gemm_wmma_f32(const float* __restrict__ A, const float* __restrict__ B,
                              const float* __restrict__ bias, float* __restrict__ C,
                              int M, int N, int K, int lda, int ldb, int ldc,
                              long long sAb, long long sBb, long long sCb,
                              int act) {
  const int wave = threadIdx.x >> 5;
  const int lane = threadIdx.x & 31;
  const int hf = lane >> 4;
  const int lr = lane & 15;
  const int ntn = (N + 15) >> 4;
  const int ntm = (M + 15) >> 4;
  const int tile = blockIdx.x * (blockDim.x >> 5) + wave;
  if (tile >= ntm * ntn) return;                 // wave-uniform exit
  const int tm = tile / ntn;
  const int tn = tile - tm * ntn;
  const float* Ab = A + (long long)blockIdx.y * sAb;
  const float* Bb = B + (long long)blockIdx.y * sBb;
  float* Cb = C + (long long)blockIdx.y * sCb;
  const int m0 = tm << 4, n0 = tn << 4;
  const int row = m0 + lr;
  const int col = n0 + lr;
  const int rowc = row < M ? row : M - 1;        // clamp: branch-free loads
  const int colc = col < N ? col : N - 1;
  const bool cok = col < N;

  const float* ap = Ab + (size_t)rowc * lda + (hf << 1);
  const float* bp = TRANSB ? (Bb + (size_t)colc * ldb + (hf << 1))
                           : (Bb + (size_t)(hf << 1) * ldb + colc);
  const int bstep = TRANSB ? 4 : 4 * ldb;

  v8f acc = {0.f, 0.f, 0.f, 0.f, 0.f, 0.f, 0.f, 0.f};
#pragma unroll 4
  for (int k0 = 0; k0 < K; k0 += 4) {
    v2f a;
    a.x = ap[0];
    a.y = ap[1];
    v2f b;
    if (TRANSB) {
      b.x = bp[0];
      b.y = bp[1];
    } else {
      b.x = bp[0];
      b.y = bp[ldb];
    }
    __builtin_prefetch((const void*)(ap + 32), 0, 3);   // global_prefetch_b8
    acc = __builtin_amdgcn_wmma_f32_16x16x4_f32(false, a, false, b, (short)0, acc,
                                                false, false);
    ap += 4;
    bp += bstep;
  }
#pragma unroll
  for (int r = 0; r < 8; ++r) {
    const int rm = m0 + r + (hf << 3);
    if (rm < M && cok) {
      float x = acc[r];
      if (bias) x += bias[col];
      Cb[(size_t)rm * ldc + col] = apply_act(x, act);
    }
  }
}

// ---------------------------------------------------------------------------
// Flash attention, one wave per 16-row query tile.
// q:(H,M,64)  k,v:(H,Nkv,64)  maskp:(Nkv)  out:(H,M,64)  scale = 64^-0.5
// Masked-out kv columns get -1e30 before the online softmax (equivalent to the
// reference's index-compaction since softmax+matmul are permutation invariant).
// K/V fragments are preloaded into distinct registers so the 16-load clauses
// overlap with the back-to-back WMMA chains instead of serializing.
// ---------------------------------------------------------------------------
__global__ void flash_attn_f32(const float* __restrict__ q, const float* __restrict__ k,
                               const float* __restrict__ v, const int* __restrict__ maskp,
                               float* __restrict__ out, int M, int Nkv) {
  __shared__ float ptile[4][16 * 17];   // per-wave padded 16x16 P tile (bank-safe)
  const int wave = threadIdx.x >> 5;
  const int lane = threadIdx.x & 31;
  const int hf = lane >> 4;
  const int lr = lane & 15;
  const int head = blockIdx.y;
  const int tile = blockIdx.x * 4 + wave;
  if ((tile << 4) >= M) return;
  const float* qh = q + (size_t)head * M * 64;
  const float* kh = k + (size_t)head * Nkv * 64;
  const float* vh = v + (size_t)head * Nkv * 64;
  float* oh = out + (size_t)head * M * 64;
  const int m0 = tile << 4;
  const int row = m0 + lr;

  v2f qreg[16];                          // whole q tile, A-frag layout
#pragma unroll
  for (int i = 0; i < 16; ++i) {
    const float* qp = qh + (size_t)row * 64 + (i << 2) + (hf << 1);
    qreg[i].x = qp[0];
    qreg[i].y = qp[1];
  }
  float mrow[8], lsum[8];
#pragma unroll
  for (int r = 0; r < 8; ++r) { mrow[r] = -1e30f; lsum[r] = 0.f; }
  v8f o0 = {0.f,0.f,0.f,0.f,0.f,0.f,0.f,0.f};
  v8f o1 = o0, o2 = o0, o3 = o0;
  float* myp = &ptile[wave][0];

  for (int n0 = 0; n0 < Nkv; n0 += 16) {
    const int col = n0 + lr;
    const float* kcol = kh + (size_t)col * 64 + (hf << 1);

    v2f kf[16];                          // distinct regs -> one load clause
#pragma unroll
    for (int i = 0; i < 16; ++i) {
      kf[i].x = kcol[i << 2];
      kf[i].y = kcol[(i << 2) + 1];
    }
    v8f s = {0.f,0.f,0.f,0.f,0.f,0.f,0.f,0.f};
#pragma unroll
    for (int i = 0; i < 16; ++i)         // S = Q * K^T (K-dim = 64)
      s = __builtin_amdgcn_wmma_f32_16x16x4_f32(false, qreg[i], false, kf[i], (short)0,
                                                s, false, false);

    const float cadd = maskp[col] ? 0.f : -1e30f;
    float sv[8], mx[8];
#pragma unroll
    for (int r = 0; r < 8; ++r) { sv[r] = s[r] * 0.125f + cadd; mx[r] = sv[r]; }
    // butterfly max, level-by-level so the 8 bpermutes per level clause together
#pragma unroll
    for (int off = 1; off < 16; off <<= 1) {
#pragma unroll
      for (int r = 0; r < 8; ++r) mx[r] = fmaxf(mx[r], __shfl_xor(mx[r], off, 32));
    }
    float nm[8], corr[8], pr[8], rs[8];
#pragma unroll
    for (int r = 0; r < 8; ++r) {
      nm[r] = fmaxf(mrow[r], mx[r]);
      corr[r] = __expf(mrow[r] - nm[r]);
      mrow[r] = nm[r];
      pr[r] = __expf(sv[r] - nm[r]);
      rs[r] = pr[r];
    }
#pragma unroll
    for (int off = 1; off < 16; off <<= 1) {
#pragma unroll
      for (int r = 0; r < 8; ++r) rs[r] += __shfl_xor(rs[r], off, 32);
    }
#pragma unroll
    for (int r = 0; r < 8; ++r) {
      lsum[r] = lsum[r] * corr[r] + rs[r];
      o0[r] *= corr[r]; o1[r] *= corr[r]; o2[r] *= corr[r]; o3[r] *= corr[r];
      myp[(r + (hf << 3)) * 17 + lr] = pr[r];   // C/D layout -> row-major in LDS
    }

    v2f af[4];                            // re-read P as A-frags (same wave, DS in-order)
#pragma unroll
    for (int j = 0; j < 4; ++j) {
      const float* pp = myp + lr * 17 + (j << 2) + (hf << 1);
      af[j].x = pp[0];
      af[j].y = pp[1];
    }
    v2f vf0[4], vf1[4], vf2[4], vf3[4];   // preload all V frags (one clause)
#pragma unroll
    for (int j = 0; j < 4; ++j) {
      const float* vp = vh + (size_t)(n0 + (j << 2) + (hf << 1)) * 64;
      vf0[j].x = vp[lr];      vf0[j].y = vp[64 + lr];
      vf1[j].x = vp[16 + lr]; vf1[j].y = vp[80 + lr];
      vf2[j].x = vp[32 + lr]; vf2[j].y = vp[96 + lr];
      vf3[j].x = vp[48 + lr]; vf3[j].y = vp[112 + lr];
    }
#pragma unroll
    for (int j = 0; j < 4; ++j) {         // O += P * V  (16x16 @ 16x64)
      o0 = __builtin_amdgcn_wmma_f32_16x16x4_f32(false, af[j], false, vf0[j], (short)0, o0, false, false);
      o1 = __builtin_amdgcn_wmma_f32_16x16x4_f32(false, af[j], false, vf1[j], (short)0, o1, false, false);
      o2 = __builtin_amdgcn_wmma_f32_16x16x4_f32(false, af[j], false, vf2[j], (short)0, o2, false, false);
      o3 = __builtin_amdgcn_wmma_f32_16x16x4_f32(false, af[j], false, vf3[j], (short)0, o3, false, false);
    }
  }
#pragma unroll
  for (int r = 0; r < 8; ++r) {
    const int rm = m0 + r + (hf << 3);
    const float inv = 1.0f / lsum[r];
    float* op = oh + (size_t)rm * 64;
    op[lr]      = o0[r] * inv;
    op[16 + lr] = o1[r] * inv;
    op[32 + lr] = o2[r] * inv;
    op[48 + lr] = o3[r] * inv;
  }
}

// --------------------------- auxiliary kernels -----------------------------

// direct 7x7 stride-2 pad-3 conv: in (64,Hin,Win) -> out (256,Ho,Wo)
__global__ void conv7x7s2(const float* __restrict__ x, const float* __restrict__ w,
                          const float* __restrict__ bias, float* __restrict__ out,
                          int Hin, int Win, int Ho, int Wo) {
  const int idx = blockIdx.x * blockDim.x + threadIdx.x;
  const int total = 256 * Ho * Wo;
  if (idx >= total) return;
  const int xo = idx % Wo;
  const int yo = (idx / Wo) % Ho;
  const int oc = idx / (Wo * Ho);
  float acc = bias[oc];
  const float* wp = w + (size_t)oc * 64 * 49;
  for (int ic = 0; ic < 64; ++ic) {
    const float* xp = x + (size_t)ic * Hin * Win;
    const float* wc = wp + ic * 49;
#pragma unroll
    for (int ky = 0; ky < 7; ++ky) {
      const int iy = yo * 2 - 3 + ky;
      if (iy < 0 || iy >= Hin) continue;
#pragma unroll
      for (int kx = 0; kx < 7; ++kx) {
        const int ix = xo * 2 - 3 + kx;
        if (ix < 0 || ix >= Win) continue;
        acc = fmaf(xp[iy * Win + ix], wc[ky * 7 + kx], acc);
      }
    }
  }
  out[idx] = acc;
}

// co(256,LW) -> fts[(head*LW+s)][c] = co[head*64+c][s]
__global__ void permute_fts(const float* __restrict__ co, float* __restrict__ fts, int LW) {
  const int idx = blockIdx.x * blockDim.x + threadIdx.x;
  const int total = 4 * LW * 64;
  if (idx >= total) return;
  const int c = idx & 63;
  const int m = idx >> 6;
  const int head = m / LW;
  const int rest = m - head * LW;
  fts[idx] = co[(size_t)(head * 64 + c) * LW + rest];
}

// query (64,HW) channel-major -> qin (HW,64) row-major
__global__ void permute_q(const float* __restrict__ query, float* __restrict__ qin, int HW) {
  const int idx = blockIdx.x * blockDim.x + threadIdx.x;
  if (idx >= HW * 64) return;
  const int c = idx & 63;
  const int m = idx >> 6;
  qin[idx] = query[(size_t)c * HW + m];
}

// projection of ref_pts -> normalized (y,x) pairs + validity mask
__global__ void ref_project(const float* __restrict__ ref_pts, const float* __restrict__ proj,
                            const float* __restrict__ pc, float* __restrict__ ref2,
                            int* __restrict__ mask) {
  const int p = blockIdx.x * blockDim.x + threadIdx.x;
  if (p >= 4096) return;
  const float r0 = ref_pts[p * 3 + 0] * (pc[3] - pc[0]) + pc[0];
  const float r1 = ref_pts[p * 3 + 1] * (pc[4] - pc[1]) + pc[1];
  const float r2 = ref_pts[p * 3 + 2] * (pc[5] - pc[2]) + pc[2];
  const float px = proj[0] * r0 + proj[1] * r1 + proj[2]  * r2 + proj[3];
  const float py = proj[4] * r0 + proj[5] * r1 + proj[6]  * r2 + proj[7];
  const float pz = proj[8] * r0 + proj[9] * r1 + proj[10] * r2 + proj[11];
  const float zc = fmaxf(pz, 1e-5f);
  const float xn = px / zc * (1.0f / 64.0f);
  const float yn = py / zc * (1.0f / 64.0f);
  const int mk = (pz > 1e-5f) && (yn > 0.f) && (yn < 1.f) && (xn < 1.f) && (xn > 0.f);
  ref2[p * 2 + 0] = yn;   // reference reverses xy -> (y, x)
  ref2[p * 2 + 1] = xn;
  mask[p] = mk;
}

// pts[head][p] = clip(ref2[p] + off[head][perm(p)], 0, 1); perm per branch
__global__ void make_pts(const float* __restrict__ ref2, const float* __restrict__ off,
                         float* __restrict__ pts, int branch) {
  const int idx = blockIdx.x * blockDim.x + threadIdx.x;
  if (idx >= 4 * 4096) return;
  const int p = idx & 4095;
  const int head = idx >> 12;
  int src;
  if (branch == 0) {
    src = p;                                            // (l,w,d) identity
  } else if (branch == 1) {
    const int a = p >> 7, bb = (p >> 2) & 31, cc = p & 3;
    src = (a * 4 + cc) * 32 + bb;                       // hd: (l,depth,w)->(l,w,depth)
  } else {
    const int d32 = p >> 7, l = (p >> 2) & 31, w = p & 3;
    src = (l * 4 + w) * 32 + d32;                       // wd: (depth,l,w)->(l,w,depth)
  }
  const int orow = head * 4096 + src;
  const float p0 = ref2[p * 2 + 0] + off[orow * 2 + 0];
  const float p1 = ref2[p * 2 + 1] + off[orow * 2 + 1];
  pts[idx * 2 + 0] = fminf(fmaxf(p0, 0.f), 1.f);
  pts[idx * 2 + 1] = fminf(fmaxf(p1, 0.f), 1.f);
}

// bilinear grid sample: img (64,64,64), pts (4*4096,2) -> ftS[(head*4096+p)][c]
__global__ void grid_sample_k(const float* __restrict__ img, const float* __restrict__ pts,
                              float* __restrict__ ftS) {
  const int idx = blockIdx.x * blockDim.x + threadIdx.x;
  if (idx >= 4 * 4096 * 64) return;
  const int c = idx & 63;
  const int hp = idx >> 6;
  const float gx = pts[hp * 2 + 0];
  const float gy = pts[hp * 2 + 1];
  const float x = (gx + 1.f) * 0.5f * 63.f;
  const float y = (gy + 1.f) * 0.5f * 63.f;
  const float x0f = floorf(x), y0f = floorf(y);
  const float wx = x - x0f, wy = y - y0f;
  const int x0 = (int)x0f, y0 = (int)y0f;
  const int x0c = min(max(x0, 0), 63), x1c = min(max(x0 + 1, 0), 63);
  const int y0c = min(max(y0, 0), 63), y1c = min(max(y0 + 1, 0), 63);
  const float* ip = img + (size_t)c * 4096;
  const float v00 = ip[y0c * 64 + x0c], v01 = ip[y0c * 64 + x1c];
  const float v10 = ip[y1c * 64 + x0c], v11 = ip[y1c * 64 + x1c];
  ftS[idx] = v00 * (1.f - wx) * (1.f - wy) + v01 * wx * (1.f - wy) +
             v10 * (1.f - wx) * wy + v11 * wx * wy;
}

// --------------------------------- host ------------------------------------

enum {
  CI_HW = 0, CI_HD, CI_WD, CI_IMG, CI_PROJ, CI_REF, CI_PC,
  CI_AHW_CW, CI_AHW_CB, CI_AHW_L1W, CI_AHW_L1B, CI_AHW_L2W, CI_AHW_L2B, CI_AHW_OW, CI_AHW_OB,
  CI_AHD_CW, CI_AHD_CB, CI_AHD_L1W, CI_AHD_L1B, CI_AHD_L2W, CI_AHD_L2B, CI_AHD_OW, CI_AHD_OB,
  CI_AWD_CW, CI_AWD_CB, CI_AWD_L1W, CI_AWD_L1B, CI_AWD_L2W, CI_AWD_L2B, CI_AWD_OW, CI_AWD_OB,
  CI_QCW, CI_QCB, CI_QL1W, CI_QL1B, CI_QL2W, CI_QL2B,
  CI_K1W, CI_K1B, CI_K2W, CI_K2B, CI_V1W, CI_V1B, CI_V2W, CI_V2B,
  CI_POW, CI_POB, CI_F1W, CI_F1B, CI_F2W, CI_F2B, CI_COUNT
};

// setup_inputs() insertion order (params flattened recursively in dict order)
static const int MAP_INS[CI_COUNT] = {
  0, 1, 2, 3, 4, 5, 6,
  7, 8, 9, 10, 11, 12, 13, 14,
  15, 16, 17, 18, 19, 20, 21, 22,
  23, 24, 25, 26, 27, 28, 29, 30,
  31, 32, 33, 34, 35, 36, 37, 38, 39, 40,
  41, 42, 43, 44, 45, 46, 47, 48, 49, 50
};
// jax pytree flattening (keys sorted at every level)
static const int MAP_SORTED[CI_COUNT] = {
  1, 0, 50, 2, 48, 49, 47,
  12, 11, 14, 13, 16, 15, 18, 17,
  4, 3, 6, 5, 8, 7, 10, 9,
  20, 19, 22, 21, 24, 23, 26, 25,
  38, 37, 40, 39, 42, 41,
  32, 31, 34, 33, 44, 43, 46, 45,
  36, 35, 28, 27, 30, 29
};

static inline void launch_gemm(hipStream_t s, const float* A, const float* B,
                               const float* bias, float* C, int M, int N, int K,
                               int lda, int ldb, int ldc,
                               long long sAb, long long sBb, long long sCb,
                               int batch, int transB, int act) {
  const int tiles = ((M + 15) / 16) * ((N + 15) / 16);
  dim3 grid((tiles + 3) / 4, batch, 1);
  if (transB)
    gemm_wmma_f32<1><<<grid, dim3(128), 0, s>>>(A, B, bias, C, M, N, K, lda, ldb, ldc,
                                                sAb, sBb, sCb, act);
  else
    gemm_wmma_f32<0><<<grid, dim3(128), 0, s>>>(A, B, bias, C, M, N, K, lda, ldb, ldc,
                                                sAb, sBb, sCb, act);
}

extern "C" void kernel_launch(void* const* d_in, const int* in_sizes, int n_in,
                              void* d_out, int out_size, void* d_ws, size_t ws_size,
                              hipStream_t stream) {
  (void)out_size; (void)ws_size;
  const bool sorted = (n_in > 0 && in_sizes[0] == 32768);  // hd first => jax-sorted
  const int* map = sorted ? MAP_SORTED : MAP_INS;
  auto IN = [&](int ci) -> const float* { return (const float*)d_in[map[ci]]; };

  // ---- workspace carving (floats, 256B aligned blocks), ~50 MB total ----
  float* W = (float*)d_ws;
  size_t ofs = 0;
  auto alloc = [&](size_t n) -> float* { float* p = W + ofs; ofs += (n + 63) & ~(size_t)63; return p; };
  float* ref2  = alloc(4096 * 2);
  int*   maskb = (int*)alloc(4096);
  float* co    = alloc(256 * 32 * 32);
  float* fts   = alloc(4096 * 64);
  float* h1    = alloc(512 * 2048);
  float* h2    = alloc(512 * 2048);
  float* offb  = alloc(16384 * 2);
  float* pts   = alloc(4 * 4096 * 2);
  float* ftS   = alloc(4 * 4096 * 64);
  float* t1    = alloc(4 * 4096 * 64);
  float* kb    = alloc(4 * 4096 * 64);
  float* vb    = alloc(4 * 4096 * 64);
  float* qin   = alloc(4096 * 64);
  float* Q0    = alloc(4096 * 256);
  float* qt    = alloc(4 * 4096 * 64);
  float* qb    = alloc(4 * 4096 * 64);
  float* ao    = alloc(4 * 4096 * 64);
  float* hocat = alloc(4096 * 256);
  float* g1    = alloc(4096 * 64);

  float* out = (float*)d_out;

  ref_project<<<dim3(16), dim3(256), 0, stream>>>(IN(CI_REF), IN(CI_PROJ), IN(CI_PC),
                                                  ref2, maskb);

  // branch tables: 0 = hw (64x64), 1 = hd (64x8), 2 = wd (64x8)
  const float* xin[3]  = { IN(CI_HW), IN(CI_HD), IN(CI_WD) };
  const int    Winb[3] = { 64, 8, 8 };
  const int    Wob[3]  = { 32, 4, 4 };
  const int    depb[3] = { 4, 32, 32 };
  const int    HWqb[3] = { 4096, 512, 512 };
  float*       outp[3] = { out, out + 262144, out + 294912 };
  const int    abase[3]= { CI_AHW_CW, CI_AHD_CW, CI_AWD_CW };

  for (int br = 0; br < 3; ++br) {
    const int Ho = 32, Wo = Wob[br], LW = Ho * Wo;
    const int depth = depb[br];
    const int Mfts = 4 * LW;           // 4096 or 512
    const int N1 = 64 * depth;         // 256 or 2048
    const int HWq = HWqb[br];
    const int ab = abase[br];
    const float *cw = IN(ab + 0), *cb = IN(ab + 1);
    const float *l1w = IN(ab + 2), *l1b = IN(ab + 3);
    const float *l2w = IN(ab + 4), *l2b = IN(ab + 5);
    const float *ow = IN(ab + 6), *ob = IN(ab + 7);

    // ---- offset branch: conv -> mlp -> tanh offsets ----
    {
      const int tot = 256 * Ho * Wo;
      conv7x7s2<<<dim3((tot + 255) / 256), dim3(256), 0, stream>>>(
          xin[br], cw, cb, co, 64, Winb[br], Ho, Wo);
    }
    permute_fts<<<dim3((Mfts * 64 + 255) / 256), dim3(256), 0, stream>>>(co, fts, LW);
    launch_gemm(stream, fts, l1w, l1b, h1, Mfts, N1, 64, 64, N1, N1, 0, 0, 0, 1, 0, ACT_LEAKY);
    launch_gemm(stream, h1, l2w, l2b, h2, Mfts, N1, N1, N1, N1, N1, 0, 0, 0, 1, 0, ACT_GELU);
    launch_gemm(stream, h2, ow, ob, offb, Mfts * depth, 2, 64, 64, 2, 2, 0, 0, 0, 1, 0, ACT_TANH);
    make_pts<<<dim3(64), dim3(256), 0, stream>>>(ref2, offb, pts, br);
    grid_sample_k<<<dim3((4 * 4096 * 64 + 255) / 256), dim3(256), 0, stream>>>(
        IN(CI_IMG), pts, ftS);

    // ---- Q path ----
    permute_q<<<dim3((HWq * 64 + 255) / 256), dim3(256), 0, stream>>>(xin[br], qin, HWq);
    launch_gemm(stream, qin, IN(CI_QCW), IN(CI_QCB), Q0, HWq, 256, 64, 64, 64, 256,
                0, 0, 0, 1, 1, ACT_NONE);                                  // qc_w^T
    launch_gemm(stream, Q0, IN(CI_QL1W), IN(CI_QL1B), qt, HWq, 64, 64, 256, 64, 64,
                64, 0, (long long)HWq * 64, 4, 0, ACT_LEAKY);
    launch_gemm(stream, qt, IN(CI_QL2W), IN(CI_QL2B), qb, HWq, 64, 64, 64, 64, 64,
                (long long)HWq * 64, 0, (long long)HWq * 64, 4, 0, ACT_GELU);

    // ---- K / V paths (batch = 4 heads, Nkv = 4096 sampled points) ----
    launch_gemm(stream, ftS, IN(CI_K1W), IN(CI_K1B), t1, 4096, 64, 64, 64, 64, 64,
                4096 * 64, 0, 4096 * 64, 4, 0, ACT_LEAKY);
    launch_gemm(stream, t1, IN(CI_K2W), IN(CI_K2B), kb, 4096, 64, 64, 64, 64, 64,
                4096 * 64, 0, 4096 * 64, 4, 0, ACT_GELU);
    launch_gemm(stream, ftS, IN(CI_V1W), IN(CI_V1B), t1, 4096, 64, 64, 64, 64, 64,
                4096 * 64, 0, 4096 * 64, 4, 0, ACT_LEAKY);
    launch_gemm(stream, t1, IN(CI_V2W), IN(CI_V2B), vb, 4096, 64, 64, 64, 64, 64,
                4096 * 64, 0, 4096 * 64, 4, 0, ACT_GELU);

    // ---- flash attention ----
    flash_attn_f32<<<dim3(HWq / 64, 4), dim3(128), 0, stream>>>(qb, kb, vb, maskb,
                                                                ao, HWq, 4096);

    // ---- output MLP: po (per head, concat heads) -> f1 -> f2 ----
    launch_gemm(stream, ao, IN(CI_POW), IN(CI_POB), hocat, HWq, 64, 64, 64, 64, 256,
                (long long)HWq * 64, 0, 64, 4, 1, ACT_NONE);               // po_w^T
    launch_gemm(stream, hocat, IN(CI_F1W), IN(CI_F1B), g1, HWq, 64, 256, 256, 64, 64,
                0, 0, 0, 1, 0, ACT_LEAKY);
    launch_gemm(stream, g1, IN(CI_F2W), IN(CI_F2B), outp[br], HWq, 64, 64, 64, 64, 64,
                0, 0, 0, 1, 0, ACT_LEAKY);
  }
}